// GGNN_19078244729004
// MI455X (gfx1250) — compile-verified
//
#include <hip/hip_runtime.h>

// Problem constants (fixed by the reference):
#define BATCH 4
#define CH    128
#define HH    128
#define WW    128
#define HWSZ  (HH * WW)          // 16384 pixels per (b, c) plane
#define EPS   0.2f
#define CGROUPS 8                // channel groups per (b,h) row in pass 3
#define CPG   (CH / CGROUPS)     // 16 channels per group

typedef __attribute__((ext_vector_type(2))) float v2f;
typedef __attribute__((ext_vector_type(8))) float v8f;

// ---------------------------------------------------------------------------
// CDNA5 async global->LDS copy (ASYNCcnt-tracked DMA), with safe fallback.
// AS(3) pointers are 32-bit; flat LDS addresses carry the LDS byte offset in
// bits [31:0] (ISA 10.2 aperture rules), so an integer-truncating cast yields
// a valid LDS pointer. Builtin params are int* (b32), per compiler feedback.
// ---------------------------------------------------------------------------
typedef __attribute__((address_space(1))) int g_i32;
typedef __attribute__((address_space(3))) int l_i32;

#if defined(__has_builtin) && __has_builtin(__builtin_amdgcn_global_load_async_to_lds_b32)
#define HAVE_ASYNC_LDS 1
__device__ __forceinline__ void async_copy_b32(const float* g, float* l) {
    __builtin_amdgcn_global_load_async_to_lds_b32(
        (g_i32*)(uintptr_t)g,
        (l_i32*)(uintptr_t)l,
        0 /*imm offset*/, 0 /*cpol*/);
}
#if __has_builtin(__builtin_amdgcn_s_wait_asynccnt)
#define WAIT_ASYNC(n) __builtin_amdgcn_s_wait_asynccnt(n)
#else
#define WAIT_ASYNC(n) asm volatile("s_wait_asynccnt %0" :: "i"(n) : "memory")
#endif
#else
#define HAVE_ASYNC_LDS 0
#endif

// ---------------------------------------------------------------------------
// Pass 1: channel mean via V_WMMA_F32_16X16X4_F32 (confirmed lowering).
// One wave32 owns 16 consecutive pixels (N columns). A = all (1/128) so
// D[m,n] = (1/128) * sum_k B[k,n]; K=4 channels per WMMA, 32 iterations.
// ---------------------------------------------------------------------------
__global__ void __launch_bounds__(256)
mean_wmma_kernel(const float* __restrict__ x, float* __restrict__ mean) {
    const int lane  = threadIdx.x & 31;
    const int wave  = threadIdx.x >> 5;
    const int gw    = blockIdx.x * 8 + wave;   // 8 waves per 256-thread block
    const int p0    = gw * 16;                 // first of 16 pixels (global id)
    const int b     = p0 / HWSZ;               // chunks never straddle a batch
    const int pix   = p0 - b * HWSZ;
    const int n     = lane & 15;
    const int khalf = (lane >> 4) & 1;         // 0 -> K={0,1}, 1 -> K={2,3}

    const float s = 1.0f / (float)CH;
    v2f a; a.x = s; a.y = s;                   // A = ones * (1/C)
    v8f acc = {};

    const float* base = x + (size_t)b * CH * HWSZ + pix + n;
    #pragma unroll 4
    for (int c = 0; c < CH; c += 4) {
        const float* p = base + (size_t)(c + 2 * khalf) * HWSZ;
        v2f bv;
        bv.x = p[0];                           // channel c + 2*khalf
        bv.y = p[HWSZ];                        // channel c + 2*khalf + 1
        acc = __builtin_amdgcn_wmma_f32_16x16x4_f32(
            false, a, false, bv, (short)0, acc, false, false);
    }
    // D VGPR0: lanes 0-15 = (M=0, N=lane) -> the 16 means.
    if (lane < 16) mean[p0 + n] = acc[0];
}

// ---------------------------------------------------------------------------
// Pass 2: 3x3 stencil on the mean map -> softmax over 9 taps.
// Center tap has g=0 -> r=5.0 which is the exact max, stable shift.
// ---------------------------------------------------------------------------
__global__ void __launch_bounds__(256)
weights_kernel(const float* __restrict__ mean, float* __restrict__ wts) {
    const int idx = blockIdx.x * blockDim.x + threadIdx.x;   // 0 .. B*HW-1
    const int b   = idx / HWSZ;
    const int pix = idx - b * HWSZ;
    const int h   = pix / WW;
    const int w   = pix - h * WW;

    const float* mb = mean + (size_t)b * HWSZ;
    const float  mc = mb[pix];

    float e[9];
    float sum = 0.0f;
    #pragma unroll
    for (int k = 0; k < 9; ++k) {
        const int di = k / 3 - 1, dj = k % 3 - 1;
        const int hh = h + di, ww2 = w + dj;
        const bool in = (hh >= 0) & (hh < HH) & (ww2 >= 0) & (ww2 < WW);
        const float mn = in ? mb[hh * WW + ww2] : 0.0f;
        const float g  = mn - mc;
        const float r  = 1.0f / (g * g + EPS);
        e[k] = __expf(r - 5.0f);
        sum += e[k];
    }
    const float inv = 1.0f / sum;
    #pragma unroll
    for (int k = 0; k < 9; ++k)
        wts[((size_t)b * 9 + k) * HWSZ + pix] = e[k] * inv;
}

// ---------------------------------------------------------------------------
// Pass 3: content-aware reassembly + 2x2 nearest replication.
// Block = one (b, h, channel-group). 128 threads, one per w.
// Async path: double-buffered global->LDS DMA of the 3 x-rows for channel
// c+1 overlapped with the 9-FMA compute + NT stores for channel c.
// Per iteration each lane issues exactly 3 async b32 copies (OOB rows are
// redirected to a dump row), so s_wait_asynccnt 3 guarantees the previous
// batch (current buffer) has landed; barrier makes it block-visible.
// Output uses non-temporal stores (134 MB stream) so it does not evict x
// from the 192 MB L2 — x's second read stays L2-resident.
// ---------------------------------------------------------------------------
__global__ void __launch_bounds__(WW)
carafe_kernel(const float* __restrict__ x, const float* __restrict__ wts,
              float* __restrict__ out) {
    const int bidx = blockIdx.x;
    const int cg   = bidx % CGROUPS;
    const int bh   = bidx / CGROUPS;
    const int h    = bh % HH;
    const int b    = bh / HH;
    const int w    = threadIdx.x;          // 0..127

    __shared__ float sw[9][WW];

    #pragma unroll
    for (int k = 0; k < 9; ++k)
        sw[k][w] = wts[((size_t)b * 9 + k) * HWSZ + h * WW + w];

    const int    c0 = cg * CPG;
    const size_t xb = (size_t)b * CH * HWSZ;
    const size_t ob = (size_t)b * CH * (4 * HWSZ);   // out plane = 256*256
    const float* xg = x + xb + (size_t)c0 * HWSZ;    // first channel of group

#if HAVE_ASYNC_LDS
    __shared__ float sx[2][3][WW + 2];
    __shared__ float dump[WW];

    // Pre-zero pad columns and (for boundary rows) the whole OOB row; the
    // async DMA only ever writes in-range rows, so no write overlap/races.
    if (w < 2) {
        #pragma unroll
        for (int u = 0; u < 2; ++u)
            #pragma unroll
            for (int r = 0; r < 3; ++r) sx[u][r][w * (WW + 1)] = 0.0f;
    }
    if (h == 0)      { sx[0][0][w + 1] = 0.0f; sx[1][0][w + 1] = 0.0f; }
    if (h == HH - 1) { sx[0][2][w + 1] = 0.0f; sx[1][2][w + 1] = 0.0f; }

    // Issue the 3-row copy for one channel into buffer `dst`.
    auto issue_rows = [&](const float* xc, float (*dst)[WW + 2]) {
        #pragma unroll
        for (int r = 0; r < 3; ++r) {
            const int  hr = h - 1 + r;
            const bool in = (hr >= 0) & (hr < HH);     // block-uniform
            const float* g = xc + (size_t)(in ? hr : h) * WW + w;
            float*       l = in ? &dst[r][w + 1] : &dump[w];
            async_copy_b32(g, l);                      // always 3 per lane
        }
    };

    issue_rows(xg, sx[0]);                              // prologue: channel c0
    for (int ci = 0; ci < CPG; ++ci) {
        const int cur = ci & 1;
        if (ci + 1 < CPG) {
            issue_rows(xg + (size_t)(ci + 1) * HWSZ, sx[cur ^ 1]);
            WAIT_ASYNC(3);      // previous batch (current buffer) complete
        } else {
            WAIT_ASYNC(0);
        }
        __syncthreads();        // buffer visible to all 4 waves

        float v = 0.0f;
        #pragma unroll
        for (int r = 0; r < 3; ++r)
            #pragma unroll
            for (int d = 0; d < 3; ++d)
                v = fmaf(sw[r * 3 + d][w], sx[cur][r][w + d], v);

        v2f v2; v2.x = v; v2.y = v;
        float* o = out + ob + (size_t)(c0 + ci) * (4 * HWSZ)
                 + (size_t)(2 * h) * (2 * WW) + 2 * w;
        __builtin_nontemporal_store(v2, (v2f*)o);              // row 2h
        __builtin_nontemporal_store(v2, (v2f*)(o + 2 * WW));   // row 2h+1
        __syncthreads();        // all reads done before buffer is re-DMAed
    }
#else
    __shared__ float sx[3][WW + 2];
    if (w < 2) {
        #pragma unroll
        for (int r = 0; r < 3; ++r) sx[r][w * (WW + 1)] = 0.0f;
    }
    __syncthreads();

    for (int ci = 0; ci < CPG; ++ci) {
        const float* xc = xg + (size_t)ci * HWSZ;
        #pragma unroll
        for (int r = 0; r < 3; ++r) {
            const int hr = h - 1 + r;
            sx[r][w + 1] = (hr >= 0 && hr < HH) ? xc[hr * WW + w] : 0.0f;
        }
        __syncthreads();

        float v = 0.0f;
        #pragma unroll
        for (int r = 0; r < 3; ++r)
            #pragma unroll
            for (int d = 0; d < 3; ++d)
                v = fmaf(sw[r * 3 + d][w], sx[r][w + d], v);

        v2f v2; v2.x = v; v2.y = v;
        float* o = out + ob + (size_t)(c0 + ci) * (4 * HWSZ)
                 + (size_t)(2 * h) * (2 * WW) + 2 * w;
        __builtin_nontemporal_store(v2, (v2f*)o);
        __builtin_nontemporal_store(v2, (v2f*)(o + 2 * WW));
        __syncthreads();
    }
#endif
}

// ---------------------------------------------------------------------------
extern "C" void kernel_launch(void* const* d_in, const int* in_sizes, int n_in,
                              void* d_out, int out_size, void* d_ws, size_t ws_size,
                              hipStream_t stream) {
    (void)in_sizes; (void)n_in; (void)out_size; (void)ws_size;
    const float* x   = (const float*)d_in[0];
    float*       out = (float*)d_out;

    float* mean = (float*)d_ws;                       // B*HW floats   (256 KB)
    float* wts  = mean + (size_t)BATCH * HWSZ;        // B*9*HW floats (2.25 MB)

    // Pass 1: 4096 waves * 16 pixels = 65536 pixels; 8 waves/block.
    mean_wmma_kernel<<<(BATCH * HWSZ / 16) / 8, 256, 0, stream>>>(x, mean);
    // Pass 2: one thread per pixel.
    weights_kernel<<<(BATCH * HWSZ) / 256, 256, 0, stream>>>(mean, wts);
    // Pass 3: (b, h, channel-group) blocks, 128 threads each.
    carafe_kernel<<<BATCH * HH * CGROUPS, WW, 0, stream>>>(x, wts, out);
}